// ApproxExp_FXP32in16out14_48644799594813
// MI455X (gfx1250) — compile-verified
//
#include <hip/hip_runtime.h>
#include <stdint.h>

// ---------------------------------------------------------------------------
// ApproxExp FXP32.16-in / FXP16.14-out via uniform-spacing LUT.
// Memory-bound streaming kernel: 8 B/elem, ~92 us floor @ 23.3 TB/s.
// CDNA5 paths used: wave32 sizing, B128 non-temporal global streaming,
// conflict-free LDS LUT (ds_load_b64), TDM tensor_load_to_lds + s_wait_tensorcnt
// for the per-block LUT broadcast.
// ---------------------------------------------------------------------------

typedef float    f32x4 __attribute__((ext_vector_type(4)));
typedef uint32_t u32x4 __attribute__((ext_vector_type(4)));
typedef int      i32x8 __attribute__((ext_vector_type(8)));
typedef int      i32x4 __attribute__((ext_vector_type(4)));

// round(exp(linspace(-10,4,17)) * 2^14)  (double-precision derivation, margins
// checked against float32 exp error; worst margin 0.047 >> f32 exp error)
__constant__ int EXPV[17] = {1, 2, 4, 10, 25, 59, 142, 340, 816, 1957,
                             4694, 11261, 27013, 64800, 155447, 372898, 894536};

#define MIN_X  (-655360)   // -10 * 2^16 (exact)
#define MAX_X  (262144)    //   4 * 2^16 (exact)
#define SEG    57344u      // 0.875 * 2^16 (exact, uniform breakpoint spacing)
#define Y_LAST 894536
#define TPB    256
#define VPT    4           // f32x4 vectors per thread

// Pack (y0, dy) pairs into global scratch so the TDM can broadcast them to LDS.
__global__ void lut_pack_kernel(int* __restrict__ g) {
  int t = threadIdx.x;
  if (t < 16) {
    g[2 * t]     = EXPV[t];
    g[2 * t + 1] = EXPV[t + 1] - EXPV[t];
  }
}

// Faithful integer pipeline.  Clamping alone reproduces the low-boundary case;
// high boundary needs one select.  t_fx = ((dx<<14)+28672)//57344 == (2dx+3)/7
// exactly.  int32 wraparound of t*dy reproduced via uint32 mul + bitcast,
// arithmetic >>14 matches jax/torch int32 semantics.
__device__ __forceinline__ int eval_fxp(int xi, const int2* __restrict__ lut) {
  int xc       = min(max(xi, MIN_X), MAX_X);          // v_med3_i32
  unsigned u   = (unsigned)(xc - MIN_X);              // 0 .. 917504
  unsigned idx = (unsigned)max((int)u - 1, 0) / SEG;  // 0 .. 15 (searchsorted-left - 1)
  int dx       = (int)(u - idx * SEG);                // 0 .. 57344
  unsigned t   = (2u * (unsigned)dx + 3u) / 7u;       // == ((dx<<14)+28672)/57344
  int2 p       = lut[idx];                            // ds_load_b64: {y0, dy}
  int prod     = (int)(t * (unsigned)p.y + 8192u);    // int32 wraparound preserved
  int interp   = p.x + (prod >> 14);
  return (xi >= MAX_X) ? Y_LAST : interp;
}

__global__ __launch_bounds__(TPB) void approx_exp_kernel(
    const float* __restrict__ x, float* __restrict__ out,
    const int* __restrict__ lut_g, long long nvec, int use_tdm) {
  __shared__ int2 sLut[16];

  if (use_tdm) {
    if (threadIdx.x < 32) {  // wave 0 only: single TDM op per block
      uint32_t lds_off = (uint32_t)(uintptr_t)(&sLut[0]);  // flat LDS addr low 32b = LDS byte offset
      uint64_t ga      = (uint64_t)(uintptr_t)lut_g;
      // D# group 0: count=1 | lds_addr | global_addr[31:0] | global_addr[56:32], type=2
      u32x4 g0 = {1u, lds_off, (uint32_t)ga,
                  (uint32_t)((ga >> 32) & 0x01FFFFFFu) | (2u << 30)};
      // D# group 1: data_size=4B; tensor_dim0=32 dwords; tensor_dim1=1;
      //             tile_dim0=32; tile_dim1/2 unused; tensor_dim0_stride=32
      i32x8 g1 = {(int)(2u << 16),   // [17:16] data_size = 2 (4 bytes)
                  (int)(32u << 16),  // [31:16] tensor_dim0 lo = 32
                  (int)(1u << 16),   // [15:0] tdim0 hi=0 | [31:16] tensor_dim1 lo = 1
                  (int)(32u << 16),  // [15:0] tdim1 hi=0 | [31:16] tile_dim0 = 32
                  0,                 // tile_dim1=0 (unused), tile_dim2=0
                  32,                // tensor_dim0_stride lo = 32
                  0, 0};
      i32x4 g2 = {0, 0, 0, 0};
      i32x4 g3 = {0, 0, 0, 0};
      i32x8 g4 = {0, 0, 0, 0, 0, 0, 0, 0};  // extra operand on clang-23 6-arg form
      __builtin_amdgcn_tensor_load_to_lds(g0, g1, g2, g3, g4, 0);
      __builtin_amdgcn_s_wait_tensorcnt(0);
    }
  } else {
    if (threadIdx.x < 16) {
      int t = threadIdx.x;
      sLut[t] = make_int2(EXPV[t], EXPV[t + 1] - EXPV[t]);
    }
  }
  __syncthreads();

  const f32x4* __restrict__ xv = (const f32x4*)x;
  f32x4* __restrict__ ov       = (f32x4*)out;
  long long base = (long long)blockIdx.x * (TPB * VPT) + threadIdx.x;

#pragma unroll
  for (int it = 0; it < VPT; ++it) {
    long long i = base + (long long)it * TPB;
    if (i < nvec) {
      f32x4 v = __builtin_nontemporal_load(&xv[i]);  // global_load_b128, TH=NT
      int a = __float2int_rn(v.x * 65536.0f);        // round-half-even, exact pow2 scale
      int b = __float2int_rn(v.y * 65536.0f);
      int c = __float2int_rn(v.z * 65536.0f);
      int d = __float2int_rn(v.w * 65536.0f);
      f32x4 r;
      r.x = (float)eval_fxp(a, sLut) * (1.0f / 16384.0f);
      r.y = (float)eval_fxp(b, sLut) * (1.0f / 16384.0f);
      r.z = (float)eval_fxp(c, sLut) * (1.0f / 16384.0f);
      r.w = (float)eval_fxp(d, sLut) * (1.0f / 16384.0f);
      __builtin_nontemporal_store(r, &ov[i]);        // global_store_b128, TH=NT
    }
  }
}

// Scalar tail (n % 4 != 0 safety; unused for the given 268M-element shape).
__global__ void approx_exp_tail_kernel(const float* __restrict__ x,
                                       float* __restrict__ out,
                                       long long start, long long n) {
  long long i = start + (long long)blockIdx.x * blockDim.x + threadIdx.x;
  if (i < n) {
    int xi       = __float2int_rn(x[i] * 65536.0f);
    int xc       = min(max(xi, MIN_X), MAX_X);
    unsigned u   = (unsigned)(xc - MIN_X);
    unsigned idx = (unsigned)max((int)u - 1, 0) / SEG;
    int dx       = (int)(u - idx * SEG);
    unsigned t   = (2u * (unsigned)dx + 3u) / 7u;
    int y0 = EXPV[idx], dy = EXPV[idx + 1] - EXPV[idx];
    int prod = (int)(t * (unsigned)dy + 8192u);
    int v    = (xi >= MAX_X) ? Y_LAST : (y0 + (prod >> 14));
    out[i]   = (float)v * (1.0f / 16384.0f);
  }
}

extern "C" void kernel_launch(void* const* d_in, const int* in_sizes, int n_in,
                              void* d_out, int out_size, void* d_ws, size_t ws_size,
                              hipStream_t stream) {
  const float* x = (const float*)d_in[0];
  float* out     = (float*)d_out;
  long long n    = (long long)in_sizes[0];
  long long nvec = n >> 2;

  int use_tdm = (d_ws != nullptr && ws_size >= 128) ? 1 : 0;
  int* lut_g  = (int*)d_ws;
  if (use_tdm) {
    lut_pack_kernel<<<1, 32, 0, stream>>>(lut_g);
  }

  if (nvec > 0) {
    long long blocks = (nvec + (long long)(TPB * VPT) - 1) / (TPB * VPT);
    approx_exp_kernel<<<(uint32_t)blocks, TPB, 0, stream>>>(x, out, lut_g, nvec, use_tdm);
  }
  long long tail = n & 3;
  if (tail) {
    approx_exp_tail_kernel<<<1, 64, 0, stream>>>(x, out, n - tail, n);
  }
}